// TableFormerAttention_84447646974323
// MI455X (gfx1250) — compile-verified
//
#include <hip/hip_runtime.h>

typedef __attribute__((ext_vector_type(2))) float v2f;
typedef __attribute__((ext_vector_type(8))) float v8f;

#define D_MODEL 4096
#define N_HEADS 32
#define HEAD_DIM 128
#define MAX_SEQ 512
#define KSPLIT 4
#define KCHUNK (D_MODEL / KSPLIT)  // 1024

__device__ __forceinline__ v8f wmma_f32_16x16x4(v2f a, v2f b, v8f c) {
    // D = A(16x4,f32) * B(4x16,f32) + C(16x16,f32)
    return __builtin_amdgcn_wmma_f32_16x16x4_f32(false, a, false, b, (short)0, c,
                                                 false, false);
}

// ---------------------------------------------------------------------------
// Split-K GEMV: part[kc][row] = sum_{k in chunk kc} W[row][k] * x[k]
// One wave computes 16 consecutive output rows over a 1024-wide K slice.
// A-tile: lanes 0-15 carry rows (K=k0,k0+1), lanes 16-31 carry (K=k0+2,k0+3)
// B-tile: x chunk broadcast across the 16 N columns -> D[m][n] == y[m] for all n
// ---------------------------------------------------------------------------
__global__ void gemv_part(const float* __restrict__ W, const float* __restrict__ x,
                          float* __restrict__ part) {
    const int wave = threadIdx.x >> 5;
    const int lane = threadIdx.x & 31;
    const int rowBase = blockIdx.x * ((int)blockDim.x >> 5) * 16 + wave * 16;
    const int kc = blockIdx.y;
    const int kstart = kc * KCHUNK;

    const int m   = lane & 15;          // row within tile
    const int kh2 = (lane >> 4) << 1;   // 0 for lanes 0-15, 2 for lanes 16-31

    const float* wrow = W + (size_t)(rowBase + m) * D_MODEL + kstart + kh2;
    const float* xp   = x + kstart + kh2;

    v8f acc = {};
#pragma unroll 8
    for (int k = 0; k < KCHUNK; k += 4) {
        v2f a = *(const v2f*)(wrow + k);   // global_load_b64, 8B-aligned
        v2f b = *(const v2f*)(xp + k);     // broadcast within half-wave
        acc = wmma_f32_16x16x4(a, b, acc);
    }

    // Column N=0 of D: lane 0 holds rows 0..7 in acc[0..7], lane 16 rows 8..15.
    if (m == 0) {
        const int r0 = rowBase + ((lane >> 4) << 3);
        float* p = part + (size_t)kc * D_MODEL;
#pragma unroll
        for (int i = 0; i < 8; ++i) p[r0 + i] = acc[i];
    }
}

__global__ void gemv_reduce(const float* __restrict__ part, const float* __restrict__ bias,
                            float* __restrict__ y) {
    const int i = blockIdx.x * blockDim.x + threadIdx.x;
    if (i < D_MODEL) {
        float s = bias[i];
#pragma unroll
        for (int c = 0; c < KSPLIT; ++c) s += part[(size_t)c * D_MODEL + i];
        y[i] = s;
    }
}

// ---------------------------------------------------------------------------
// KV cache passthrough copy with row `step` replaced by the new k/v vectors.
// Layout: [h][s][d], flat float4 grid-stride (16 MB read + 16 MB write).
// ---------------------------------------------------------------------------
__global__ void kv_update(const float* __restrict__ kin, const float* __restrict__ vin,
                          const float* __restrict__ knew, const float* __restrict__ vnew,
                          const int* __restrict__ step_p,
                          float* __restrict__ kout, float* __restrict__ vout) {
    const int step = *step_p;
    const size_t total = (size_t)N_HEADS * MAX_SEQ * HEAD_DIM / 4;  // float4 count
    for (size_t i = (size_t)blockIdx.x * blockDim.x + threadIdx.x; i < total;
         i += (size_t)gridDim.x * blockDim.x) {
        const size_t e = i * 4;                    // element index
        const int s = (int)((e >> 7) & (MAX_SEQ - 1));
        float4 kv = ((const float4*)kin)[i];
        float4 vv = ((const float4*)vin)[i];
        if (s == step) {
            const int h = (int)(e >> 16);          // e / (MAX_SEQ*HEAD_DIM)
            const int d = (int)(e & (HEAD_DIM - 1));
            kv = *(const float4*)(knew + h * HEAD_DIM + d);
            vv = *(const float4*)(vnew + h * HEAD_DIM + d);
        }
        ((float4*)kout)[i] = kv;
        ((float4*)vout)[i] = vv;
    }
}

// ---------------------------------------------------------------------------
// Per-head attention: scores = (q . K_s)*scale (masked), softmax, out = w^T V.
// One workgroup (128 threads = 4 waves) per head.
// ---------------------------------------------------------------------------
__global__ void attn_kernel(const float* __restrict__ q,
                            const float* __restrict__ kbuf,
                            const float* __restrict__ vbuf,
                            const int* __restrict__ step_p,
                            float* __restrict__ attn_out) {
    __shared__ float sc[MAX_SEQ];
    __shared__ float red[128];

    const int h    = blockIdx.x;
    const int tid  = threadIdx.x;
    const int wave = tid >> 5;
    const int lane = tid & 31;
    const int step = *step_p;
    const float scale = 0.08838834764831845f;  // 1/sqrt(128)

    const float* Kh = kbuf + (size_t)h * MAX_SEQ * HEAD_DIM;
    const float* Vh = vbuf + (size_t)h * MAX_SEQ * HEAD_DIM;
    const float* qh = q + h * HEAD_DIM;

    const int m   = lane & 15;
    const int kh2 = (lane >> 4) << 1;

    // ---- Phase 1: scores for 16 seq positions per WMMA tile ----
    for (int tile = wave; tile < MAX_SEQ / 16; tile += 4) {
        const int sbase = tile << 4;
        const float* krow = Kh + (size_t)(sbase + m) * HEAD_DIM + kh2;
        const float* qp   = qh + kh2;
        v8f acc = {};
#pragma unroll
        for (int k = 0; k < HEAD_DIM; k += 4) {
            v2f a = *(const v2f*)(krow + k);
            v2f b = *(const v2f*)(qp + k);
            acc = wmma_f32_16x16x4(a, b, acc);
        }
        if (m == 0) {
            const int s0 = sbase + ((lane >> 4) << 3);
#pragma unroll
            for (int i = 0; i < 8; ++i) {
                const int s = s0 + i;
                sc[s] = (s <= step) ? acc[i] * scale : -INFINITY;
            }
        }
    }
    __syncthreads();

    // ---- Phase 2: softmax over MAX_SEQ in LDS ----
    float lmax = -INFINITY;
    for (int s = tid; s < MAX_SEQ; s += 128) lmax = fmaxf(lmax, sc[s]);
    red[tid] = lmax;
    __syncthreads();
    for (int off = 64; off > 0; off >>= 1) {
        if (tid < off) red[tid] = fmaxf(red[tid], red[tid + off]);
        __syncthreads();
    }
    const float gmax = red[0];
    __syncthreads();

    float lsum = 0.0f;
    for (int s = tid; s < MAX_SEQ; s += 128) {
        const float e = __expf(sc[s] - gmax);   // exp(-inf - gmax) == 0 beyond step
        sc[s] = e;
        lsum += e;
    }
    red[tid] = lsum;
    __syncthreads();
    for (int off = 64; off > 0; off >>= 1) {
        if (tid < off) red[tid] += red[tid + off];
        __syncthreads();
    }
    const float inv = 1.0f / red[0];
    __syncthreads();

    // ---- Phase 3: out[d] = sum_s w[s] * V[s][d], 16 d-columns per tile ----
    const int s_end = (step + 4) & ~3;  // round (step+1) up to multiple of 4, <= 512
    const int n = lane & 15;
    for (int dtile = wave; dtile < HEAD_DIM / 16; dtile += 4) {
        const int dbase = dtile << 4;
        const float* vcol = Vh + dbase + n;
        v8f acc = {};
        for (int s0 = 0; s0 < s_end; s0 += 4) {
            v2f a, b;
            a.x = sc[s0 + kh2];                                 // broadcast weights
            a.y = sc[s0 + kh2 + 1];
            b.x = vcol[(size_t)(s0 + kh2) * HEAD_DIM];          // V[s][dbase+n]
            b.y = vcol[(size_t)(s0 + kh2 + 1) * HEAD_DIM];
            acc = wmma_f32_16x16x4(a, b, acc);
        }
        if (lane < 16) attn_out[h * HEAD_DIM + dbase + n] = acc[0] * inv;
    }
}

// ---------------------------------------------------------------------------
extern "C" void kernel_launch(void* const* d_in, const int* in_sizes, int n_in,
                              void* d_out, int out_size, void* d_ws, size_t ws_size,
                              hipStream_t stream) {
    (void)in_sizes; (void)n_in; (void)out_size; (void)ws_size;

    const float* query = (const float*)d_in[0];
    const float* kvsrc = (const float*)d_in[1];
    const int*   step  = (const int*)d_in[2];
    const float* kbuf  = (const float*)d_in[3];
    const float* vbuf  = (const float*)d_in[4];
    const float* Wq = (const float*)d_in[5];
    const float* bq = (const float*)d_in[6];
    const float* Wk = (const float*)d_in[7];
    const float* bk = (const float*)d_in[8];
    const float* Wv = (const float*)d_in[9];
    const float* bv = (const float*)d_in[10];
    const float* Wo = (const float*)d_in[11];
    const float* bo = (const float*)d_in[12];

    float* out  = (float*)d_out;                       // [4096]
    float* kout = out + D_MODEL;                       // [32*512*128]
    float* vout = kout + (size_t)N_HEADS * MAX_SEQ * HEAD_DIM;

    float* ws   = (float*)d_ws;                        // 8*4096 floats = 128 KB
    float* qv   = ws;
    float* knew = ws + D_MODEL;
    float* vnew = ws + 2 * D_MODEL;
    float* attn = ws + 3 * D_MODEL;
    float* part = ws + 4 * D_MODEL;                    // KSPLIT * D_MODEL

    const dim3 gg(D_MODEL / 128, KSPLIT);              // 8 waves/block * 16 rows/wave

    gemv_part<<<gg, 256, 0, stream>>>(Wq, query, part);
    gemv_reduce<<<D_MODEL / 256, 256, 0, stream>>>(part, bq, qv);
    gemv_part<<<gg, 256, 0, stream>>>(Wk, kvsrc, part);
    gemv_reduce<<<D_MODEL / 256, 256, 0, stream>>>(part, bk, knew);
    gemv_part<<<gg, 256, 0, stream>>>(Wv, kvsrc, part);
    gemv_reduce<<<D_MODEL / 256, 256, 0, stream>>>(part, bv, vnew);

    kv_update<<<1024, 256, 0, stream>>>(kbuf, vbuf, knew, vnew, step, kout, vout);

    attn_kernel<<<N_HEADS, 128, 0, stream>>>(qv, kout, vout, step, attn);

    gemv_part<<<gg, 256, 0, stream>>>(Wo, attn, part);
    gemv_reduce<<<D_MODEL / 256, 256, 0, stream>>>(part, bo, out);
}